// KANStressPredictor_36344013259139
// MI455X (gfx1250) — compile-verified
//
#include <hip/hip_runtime.h>

// KANStressPredictor elementwise kernel for gfx1250 (MI455X).
// Memory-bound streaming map: [B,T,3] f32 -> [B,T,3] f32.
// 4 points per thread => 3x b128 loads + 3x b128 stores, all non-temporal.

typedef float v4f __attribute__((ext_vector_type(4)));

__device__ __forceinline__ void kan_point(float s0, float s1, float s2,
                                          float ki0, float ki1,
                                          float& o0, float& o1, float& o2) {
    // C = 2E + I (symmetric 2x2): c00, c11, c01 = s2
    const float c00  = fmaf(2.0f, s0, 1.0f);
    const float c11  = fmaf(2.0f, s1, 1.0f);
    const float mean = 0.5f * (c00 + c11);
    const float hd   = 0.5f * (c00 - c11);
    // rad = sqrt(0.25*(c00-c11)^2 + c01^2)
    const float rad  = __builtin_amdgcn_sqrtf(fmaf(hd, hd, s2 * s2));
    const float sq0  = mean - rad;            // ascending eigenvalues of C
    const float sq1  = mean + rad;

    // log2-domain fusion:
    //   det(C) = sq0*sq1  =>  log2(det) = l0 + l1
    //   log(J) = 0.5*ln(det) = 0.5*ln2*(l0+l1)
    //   reg_eig_k^ki0 = (sqrt(sq_k) * det^(-1/6))^ki0
    //                 = exp2( ki0 * (0.5*l_k - (l0+l1)/6) )
    const float l0 = __builtin_amdgcn_logf(sq0);
    const float l1 = __builtin_amdgcn_logf(sq1);
    const float L  = l0 + l1;                 // log2(det)
    const float t  = L * (1.0f / 6.0f);

    o0 = __builtin_amdgcn_exp2f(ki0 * fmaf(0.5f, l0, -t));
    o1 = __builtin_amdgcn_exp2f(ki0 * fmaf(0.5f, l1, -t));
    o2 = ki1 * (0.34657359027997264f * L);    // 0.5*ln2 * log2(det) = log(J)
}

__global__ __launch_bounds__(256) void kan_stress_vec4(
    const float* __restrict__ strain,
    const float* __restrict__ ki0p,
    const float* __restrict__ ki1p,
    float* __restrict__ out,
    int n4) {
    const int i = blockIdx.x * blockDim.x + threadIdx.x;
    if (i >= n4) return;

    const float ki0 = ki0p[0];   // uniform -> s_load_b32
    const float ki1 = ki1p[0];

    // 4 points = 12 floats = 3 x 128-bit, single-pass stream: NT hints so the
    // 201MB working set (> 192MB L2) doesn't thrash the cache.
    const v4f* ip = (const v4f*)strain + (size_t)i * 3;
    const v4f a = __builtin_nontemporal_load(ip + 0);
    const v4f b = __builtin_nontemporal_load(ip + 1);
    const v4f c = __builtin_nontemporal_load(ip + 2);

    const float s[12] = {a.x, a.y, a.z, a.w,
                         b.x, b.y, b.z, b.w,
                         c.x, c.y, c.z, c.w};
    float r[12];
#pragma unroll
    for (int p = 0; p < 4; ++p) {
        kan_point(s[3 * p + 0], s[3 * p + 1], s[3 * p + 2], ki0, ki1,
                  r[3 * p + 0], r[3 * p + 1], r[3 * p + 2]);
    }

    v4f* op = (v4f*)out + (size_t)i * 3;
    const v4f r0 = {r[0], r[1], r[2],  r[3]};
    const v4f r1 = {r[4], r[5], r[6],  r[7]};
    const v4f r2 = {r[8], r[9], r[10], r[11]};
    __builtin_nontemporal_store(r0, op + 0);
    __builtin_nontemporal_store(r1, op + 1);
    __builtin_nontemporal_store(r2, op + 2);
}

// Scalar tail for point counts not divisible by 4 (empty for 4096x2048).
__global__ __launch_bounds__(256) void kan_stress_tail(
    const float* __restrict__ strain,
    const float* __restrict__ ki0p,
    const float* __restrict__ ki1p,
    float* __restrict__ out,
    int start, int n_points) {
    const int i = start + blockIdx.x * blockDim.x + threadIdx.x;
    if (i >= n_points) return;
    const float ki0 = ki0p[0];
    const float ki1 = ki1p[0];
    const size_t base = (size_t)i * 3;
    float o0, o1, o2;
    kan_point(strain[base + 0], strain[base + 1], strain[base + 2], ki0, ki1,
              o0, o1, o2);
    out[base + 0] = o0;
    out[base + 1] = o1;
    out[base + 2] = o2;
}

extern "C" void kernel_launch(void* const* d_in, const int* in_sizes, int n_in,
                              void* d_out, int out_size, void* d_ws, size_t ws_size,
                              hipStream_t stream) {
    (void)n_in; (void)out_size; (void)d_ws; (void)ws_size;
    const float* strain = (const float*)d_in[0];
    const float* ki0    = (const float*)d_in[1];
    const float* ki1    = (const float*)d_in[2];
    float* out          = (float*)d_out;

    const long long n_points = (long long)in_sizes[0] / 3;   // B*T
    const int n4  = (int)(n_points / 4);                     // float4-triplet groups
    const int rem = (int)(n_points - (long long)n4 * 4);

    if (n4 > 0) {
        const int block = 256;
        const int grid  = (n4 + block - 1) / block;
        kan_stress_vec4<<<grid, block, 0, stream>>>(strain, ki0, ki1, out, n4);
    }
    if (rem > 0) {
        kan_stress_tail<<<1, 256, 0, stream>>>(strain, ki0, ki1, out,
                                               n4 * 4, (int)n_points);
    }
}